// HydrologicalModel_53609781789184
// MI455X (gfx1250) — compile-verified
//
#include <hip/hip_runtime.h>
#include <cstdint>

// Problem geometry (fixed by the reference)
#define T_STEPS 2048
#define SITES   2048            // B*H = 32*64
#define LANES   32              // one wave32 per block, one site per lane
#define CHUNK   64              // timesteps staged per LDS buffer
#define NCHUNK  (T_STEPS / CHUNK)
#define ROW_BYTES   (SITES * 2 * 4)          // bytes per timestep of forcings (16 KB)
#define CHUNK_BYTES (CHUNK * LANES * 2 * 4)  // LDS bytes per chunk buffer (16 KB)

__launch_bounds__(LANES, 1)
__global__ void hydro_scan_kernel(const float* __restrict__ forcings,
                                  const float* __restrict__ states0,
                                  const float* __restrict__ params,
                                  float* __restrict__ out)
{
    // Double-buffered staging of forcings: 2 x 16 KB (WGP has 320 KB)
    __shared__ __align__(16) float smem[2 * CHUNK * LANES * 2];

    const int lane  = (int)threadIdx.x;        // 0..31
    const int site0 = (int)blockIdx.x * LANES; // this wave's site base
    const int site  = site0 + lane;

    // ---- static parameter transform: min + (max-min)*u ----
    const float4 up = ((const float4*)params)[site];
    const float smax = 10.0f  + 490.0f  * up.x;
    const float k1   = 0.01f  + 0.89f   * up.y;
    const float k2   = 0.001f + 0.199f  * up.z;
    const float kb   = 0.001f + 0.099f  * up.w;
    const float inv_smax = 1.0f / smax;     // hoist the divide off the chain
    const float c1 = 1.0f - k1 - k2;        // S1' = max(fma(S1,c1,P) - et, 0)
    const float cb = 1.0f - kb;             // S2' = max(fma(S2,cb,perc), 0)

    const float2 s0 = ((const float2*)states0)[site];
    float S1 = s0.x, S2 = s0.y;

    // d_out = fluxes [T,B,H,4] then states [T,B,H,2], concatenated flat
    float4* __restrict__ flux_out  = (float4*)out;
    float2* __restrict__ state_out = (float2*)(out + (size_t)T_STEPS * SITES * 4);

    // ---- async global->LDS copy geometry ----
    // Per async b128 instruction: lanes 0..15 move timestep-row r (256 B),
    // lanes 16..31 move row r+1. 32 instructions cover one 64-step chunk.
    const uint32_t col       = (uint32_t)((lane & 15) * 16);   // byte within 256 B row
    const uint32_t rsel      = (uint32_t)(lane >> 4);          // 0 or 1
    const uint32_t lane_goff = rsel * (uint32_t)ROW_BYTES + (uint32_t)site0 * 8u + col;
    const uint32_t lane_loff = rsel * (LANES * 8u) + col;      // within LDS chunk (256 B rows)
    const uint32_t lds_base  = (uint32_t)(uintptr_t)(&smem[0]);
    const uint64_t gbase     = (uint64_t)(uintptr_t)forcings;

    auto prefetch_chunk = [&](int c) {
        uint32_t goff = (uint32_t)c * (uint32_t)CHUNK * (uint32_t)ROW_BYTES + lane_goff;
        uint32_t loff = lds_base + (uint32_t)(c & 1) * (uint32_t)CHUNK_BYTES + lane_loff;
        #pragma unroll
        for (int j = 0; j < CHUNK / 2; ++j) {
            // GVS mode: 64-bit SGPR base + 32-bit per-lane offset; VDST = LDS byte addr.
            asm volatile("global_load_async_to_lds_b128 %0, %1, %2"
                         :
                         : "v"(loff), "v"(goff), "s"(gbase)
                         : "memory");
            goff += 2u * (uint32_t)ROW_BYTES;
            loff += 2u * (LANES * 8u);
        }
    };

    prefetch_chunk(0);

    for (int c = 0; c < NCHUNK; ++c) {
        if (c + 1 < NCHUNK) {
            prefetch_chunk(c + 1);                       // keep the async engine busy
            // 32 newly-issued ops may remain; in-order completion guarantees
            // chunk c's 32 transfers have landed once ASYNCcnt <= 32.
            asm volatile("s_wait_asynccnt 32" ::: "memory");
        } else {
            asm volatile("s_wait_asynccnt 0" ::: "memory");
        }

        const float2* __restrict__ sbuf =
            (const float2*)(&smem[(c & 1) * CHUNK * LANES * 2]);

        size_t oidx = (size_t)(c * CHUNK) * SITES + (size_t)site;
        #pragma unroll 16
        for (int tl = 0; tl < CHUNK; ++tl) {
            const float2 f = sbuf[tl * LANES + lane];   // ds_load_b64, conflict-free
            const float P = f.x, PET = f.y;

            // Off-chain prep from loaded forcing (PET >= 0 by construction):
            const float w = PET * inv_smax;

            // 4-deep state chain:
            //   z = S1*w  ||  u = fma(S1,c1,P)
            //   et = clamp(z, 0, PET)           (v_med3)
            //   S1' = max(u - et, 0)
            const float z  = S1 * w;
            const float u  = __builtin_fmaf(S1, c1, P);
            const float et = fminf(fmaxf(z, 0.0f), PET);   // == PET*clip(S1/smax,0,1)

            // Fluxes (all off the critical path)
            const float q1v   = k1 * S1;
            const float percv = k2 * S1;
            const float qbv   = kb * S2;
            flux_out[oidx] = make_float4(et, q1v, percv, qbv);   // b128, coalesced

            S1 = fmaxf(u - et, 0.0f);
            S2 = fmaxf(__builtin_fmaf(S2, cb, percv), 0.0f);     // 2-deep, parallel
            state_out[oidx] = make_float2(S1, S2);               // b64, coalesced

            oidx += SITES;
        }
    }
}

extern "C" void kernel_launch(void* const* d_in, const int* in_sizes, int n_in,
                              void* d_out, int out_size, void* d_ws, size_t ws_size,
                              hipStream_t stream) {
    const float* forcings = (const float*)d_in[0];   // [T,B,H,2]
    const float* states   = (const float*)d_in[1];   // [B,H,2]
    const float* params   = (const float*)d_in[2];   // [B,H,4]
    hydro_scan_kernel<<<SITES / LANES, LANES, 0, stream>>>(
        forcings, states, params, (float*)d_out);
}